// GTN_49005576847713
// MI455X (gfx1250) — compile-verified
//
#include <hip/hip_runtime.h>
#include <hip/hip_bf16.h>

#define NN 2048          // graph size
#define EE 5             // edge types
#define CC 2             // channels
#define FF 512           // feature dim
#define DD 64            // hidden dim
#define OO 16            // output dim
#define MM 512           // gathered nodes

typedef float v2f __attribute__((ext_vector_type(2)));
typedef float v8f __attribute__((ext_vector_type(8)));

// ---------------------------------------------------------------------------
// Tiny softmax over E for the three weight tensors -> 30 floats in ws
// ---------------------------------------------------------------------------
__global__ void softmax_weights_k(const float* __restrict__ W1_0,
                                  const float* __restrict__ W2_0,
                                  const float* __restrict__ W1_1,
                                  float* __restrict__ wsm) {
    if (threadIdx.x == 0 && blockIdx.x == 0) {
        const float* src[3] = {W1_0, W2_0, W1_1};
        for (int t = 0; t < 3; ++t) {
            for (int c = 0; c < CC; ++c) {
                float mx = -1e30f;
                for (int e = 0; e < EE; ++e) mx = fmaxf(mx, src[t][c * EE + e]);
                float s = 0.f;
                for (int e = 0; e < EE; ++e) s += expf(src[t][c * EE + e] - mx);
                for (int e = 0; e < EE; ++e)
                    wsm[t * (CC * EE) + c * EE + e] = expf(src[t][c * EE + e] - mx) / s;
            }
        }
    }
}

// ---------------------------------------------------------------------------
// Q[c] = sum_e w[c,e] * A[e]   (both channels in one pass over A)
// ---------------------------------------------------------------------------
__global__ __launch_bounds__(256)
void wsum2ch_k(const float* __restrict__ A, const float* __restrict__ w2e,
               float* __restrict__ Q) {
    __shared__ float w[CC * EE];
    if (threadIdx.x < CC * EE) w[threadIdx.x] = w2e[threadIdx.x];
    __syncthreads();
    const size_t NNt = (size_t)NN * NN;
    size_t i = (size_t)blockIdx.x * 256 + threadIdx.x;   // grid covers NN*NN exactly
    float a0 = A[i], a1 = A[NNt + i], a2 = A[2 * NNt + i],
          a3 = A[3 * NNt + i], a4 = A[4 * NNt + i];
    Q[i]       = a0 * w[0] + a1 * w[1] + a2 * w[2] + a3 * w[3] + a4 * w[4];
    Q[NNt + i] = a0 * w[5] + a1 * w[6] + a2 * w[7] + a3 * w[8] + a4 * w[9];
}

// ---------------------------------------------------------------------------
// Batched fp32 GEMM  C[ch] = A[ch] @ B[ch],  2048^3, via V_WMMA_F32_16X16X4_F32
// block = 256 threads = 8 waves (4 x 2) -> 128x128 block tile; 32x64 per wave:
// 2 A-frags x 4 B-frags = 8 WMMAs per K-step of 4 (10 vmem : 8 wmma).
// No divergence: EXEC stays all-ones as WMMA requires.
// ---------------------------------------------------------------------------
__global__ __launch_bounds__(256)
void gemm_f32_wmma_k(const float* __restrict__ Ag, const float* __restrict__ Bg,
                     float* __restrict__ Cg) {
    const int lane  = threadIdx.x & 31;
    const int wave  = threadIdx.x >> 5;   // 0..7
    const int waveM = wave >> 1;          // 0..3  (32-row strips)
    const int waveN = wave & 1;           // 0..1  (64-col strips)
    const int ch    = blockIdx.z;

    const float* A = Ag + (size_t)ch * NN * NN;
    const float* B = Bg + (size_t)ch * NN * NN;
    float*       C = Cg + (size_t)ch * NN * NN;

    const int m0 = blockIdx.y * 128 + waveM * 32;
    const int n0 = blockIdx.x * 128 + waveN * 64;

    const int ml = lane & 15;   // M (for A frag) / N (for B frag) within 16-tile
    const int hi = lane >> 4;   // selects K pair {0,1} vs {2,3}

    v8f acc[2][4];
#pragma unroll
    for (int i = 0; i < 2; ++i)
#pragma unroll
        for (int j = 0; j < 4; ++j) acc[i][j] = (v8f){};

    const float* Arow0 = A + (size_t)(m0 + ml) * NN;        // M tile 0
    const float* Arow1 = A + (size_t)(m0 + 16 + ml) * NN;   // M tile 1

    for (int kb = 0; kb < NN; kb += 4) {
        const int k0 = kb + 2 * hi;
        // A 16x4 frags: lane holds (row=ml(+16), k0..k0+1) -> aligned float2
        v2f a0 = *(const v2f*)(Arow0 + k0);
        v2f a1 = *(const v2f*)(Arow1 + k0);
        // B 4x16 frags: lane holds (k0..k0+1, col=n0+16j+ml), coalesced over lanes
        v2f b[4];
        const float* Brow0 = B + (size_t)k0 * NN + n0 + ml;
        const float* Brow1 = B + (size_t)(k0 + 1) * NN + n0 + ml;
#pragma unroll
        for (int j = 0; j < 4; ++j) {
            b[j].x = Brow0[16 * j];
            b[j].y = Brow1[16 * j];
        }
#pragma unroll
        for (int j = 0; j < 4; ++j) {
            acc[0][j] = __builtin_amdgcn_wmma_f32_16x16x4_f32(false, a0, false, b[j],
                                                              (short)0, acc[0][j], false, false);
            acc[1][j] = __builtin_amdgcn_wmma_f32_16x16x4_f32(false, a1, false, b[j],
                                                              (short)0, acc[1][j], false, false);
        }
    }

    // D layout: VGPR r -> row r + 8*hi, col = ml (per 16x16 tile); coalesced stores
    const int cn = n0 + ml;
#pragma unroll
    for (int i = 0; i < 2; ++i) {
#pragma unroll
        for (int r = 0; r < 8; ++r) {
            const size_t row = (size_t)(m0 + 16 * i + r + 8 * hi) * NN;
#pragma unroll
            for (int j = 0; j < 4; ++j)
                C[row + cn + 16 * j] = acc[i][j][r];
        }
    }
}

// ---------------------------------------------------------------------------
// Column normalization: d[j] = colsum_j - H[j,j] + 1 ; dinv = (d==0) ? 0 : 1/d
// ---------------------------------------------------------------------------
__global__ __launch_bounds__(256)
void colsum_k(const float* __restrict__ H, float* __restrict__ dinv) {
    const int ch = blockIdx.y;
    const float* Hc = H + (size_t)ch * NN * NN;
    const int j = blockIdx.x * 256 + threadIdx.x;
    float s = 0.f;
    for (int i = 0; i < NN; ++i) s += Hc[(size_t)i * NN + j];   // coalesced over j
    s = s - Hc[(size_t)j * NN + j] + 1.0f;
    dinv[ch * NN + j] = (s == 0.f) ? 0.f : 1.0f / s;
}

__global__ __launch_bounds__(256)
void norm_scale_k(const float* __restrict__ H, const float* __restrict__ dinv,
                  float* __restrict__ Hn) {
    const int ch = blockIdx.y;
    const float* Hc = H + (size_t)ch * NN * NN;
    float* Hnc = Hn + (size_t)ch * NN * NN;
    const float* dv = dinv + ch * NN;
    const size_t idx = (size_t)blockIdx.x * 256 + threadIdx.x;  // grid covers NN*NN
    const int j = (int)(idx & (NN - 1));
    const int i = (int)(idx >> 11);
    const float v = (i == j) ? 1.0f : Hc[idx];
    Hnc[idx] = v * dv[j];
}

// ---------------------------------------------------------------------------
// XW = features @ weight        [2048,512]x[512,64]
// ---------------------------------------------------------------------------
__global__ __launch_bounds__(DD)
void xw_k(const float* __restrict__ X, const float* __restrict__ W,
          float* __restrict__ XW) {
    const int n = blockIdx.x, d = threadIdx.x;
    float s = 0.f;
    for (int f = 0; f < FF; ++f) s += X[n * FF + f] * W[f * DD + d];
    XW[n * DD + d] = s;
}

// ---------------------------------------------------------------------------
// Xc[c,m,d] = relu( sum_n Hn[c,n,m] * XW[n,d] )
// ---------------------------------------------------------------------------
__global__ __launch_bounds__(DD)
void xc_k(const float* __restrict__ Hn, const float* __restrict__ XW,
          float* __restrict__ Xc) {
    const int m = blockIdx.x, ch = blockIdx.y, d = threadIdx.x;
    const float* Hc = Hn + (size_t)ch * NN * NN;
    float s = 0.f;
    for (int n = 0; n < NN; ++n) s += Hc[(size_t)n * NN + m] * XW[n * DD + d];
    Xc[((size_t)ch * NN + m) * DD + d] = fmaxf(s, 0.f);
}

// ---------------------------------------------------------------------------
// Y[n,o] = relu( lin1_b[o] + sum_{c,d} Xc[c,n,d] * lin1_w[o, c*DD+d] )
// ---------------------------------------------------------------------------
__global__ __launch_bounds__(DD)
void y_k(const float* __restrict__ Xc, const float* __restrict__ W,
         const float* __restrict__ b, float* __restrict__ Y) {
    const int n = blockIdx.x, o = threadIdx.x;
    float s = b[o];
    for (int c = 0; c < CC; ++c)
        for (int d = 0; d < DD; ++d)
            s += Xc[((size_t)c * NN + n) * DD + d] * W[o * (CC * DD) + c * DD + d];
    Y[n * DD + o] = fmaxf(s, 0.f);
}

// ---------------------------------------------------------------------------
// predict[m,o] = lin2_b[o] + sum_d Y[nodes[m],d] * lin2_w[o,d]
// ---------------------------------------------------------------------------
__global__ __launch_bounds__(256)
void predict_k(const float* __restrict__ Y, const int* __restrict__ nodes,
               const float* __restrict__ W, const float* __restrict__ b,
               float* __restrict__ out) {
    const int idx = blockIdx.x * 256 + threadIdx.x;   // MM*OO = 8192
    const int m = idx >> 4, o = idx & 15;
    const int node = nodes[m];
    float s = b[o];
    for (int d = 0; d < DD; ++d) s += Y[node * DD + d] * W[o * DD + d];
    out[idx] = s;
}

// ---------------------------------------------------------------------------
extern "C" void kernel_launch(void* const* d_in, const int* in_sizes, int n_in,
                              void* d_out, int out_size, void* d_ws, size_t ws_size,
                              hipStream_t stream) {
    const float* A        = (const float*)d_in[0];
    const float* features = (const float*)d_in[1];
    const int*   nodes    = (const int*)d_in[2];
    const float* W1_0     = (const float*)d_in[3];
    const float* W2_0     = (const float*)d_in[4];
    const float* W1_1     = (const float*)d_in[5];
    const float* weight   = (const float*)d_in[6];
    const float* lin1_w   = (const float*)d_in[7];
    const float* lin1_b   = (const float*)d_in[8];
    const float* lin2_w   = (const float*)d_in[9];
    const float* lin2_b   = (const float*)d_in[10];

    // workspace layout (floats); 3 big C*N*N slots + small buffers (~103 MB)
    const size_t SLOT = (size_t)CC * NN * NN;
    float* ws    = (float*)d_ws;
    float* slot0 = ws;                 // Q1 -> Hn -> Hn2
    float* slot1 = ws + SLOT;          // Q2 -> Q1b
    float* slot2 = ws + 2 * SLOT;      // H  -> H2
    float* wsm   = ws + 3 * SLOT;      // 30 softmax weights
    float* dinv  = wsm + 64;           // CC*NN
    float* XW    = dinv + (size_t)CC * NN;            // NN*DD
    float* Xc    = XW + (size_t)NN * DD;              // CC*NN*DD
    float* Yb    = Xc + (size_t)CC * NN * DD;         // NN*DD

    const int elemBlocks = (NN * NN) / 256;           // 16384
    const dim3 gemmGrid(NN / 128, NN / 128, CC);      // (16,16,2)

    // softmax mixes
    softmax_weights_k<<<1, 32, 0, stream>>>(W1_0, W2_0, W1_1, wsm);

    // Q1, Q2
    wsum2ch_k<<<elemBlocks, 256, 0, stream>>>(A, wsm + 0,  slot0);
    wsum2ch_k<<<elemBlocks, 256, 0, stream>>>(A, wsm + 10, slot1);

    // H = Q1 @ Q2
    gemm_f32_wmma_k<<<gemmGrid, 256, 0, stream>>>(slot0, slot1, slot2);

    // Hn = norm(H)   (into slot0)
    colsum_k<<<dim3(NN / 256, CC), 256, 0, stream>>>(slot2, dinv);
    norm_scale_k<<<dim3(elemBlocks, CC), 256, 0, stream>>>(slot2, dinv, slot0);

    // Q1b (into slot1), H2 = Hn @ Q1b (into slot2)
    wsum2ch_k<<<elemBlocks, 256, 0, stream>>>(A, wsm + 20, slot1);
    gemm_f32_wmma_k<<<gemmGrid, 256, 0, stream>>>(slot0, slot1, slot2);

    // Hn2 = norm(H2) (into slot0)
    colsum_k<<<dim3(NN / 256, CC), 256, 0, stream>>>(slot2, dinv);
    norm_scale_k<<<dim3(elemBlocks, CC), 256, 0, stream>>>(slot2, dinv, slot0);

    // tail: XW, Xc, Y, predict
    xw_k<<<NN, DD, 0, stream>>>(features, weight, XW);
    xc_k<<<dim3(NN, CC), DD, 0, stream>>>(slot0, XW, Xc);
    y_k<<<NN, DD, 0, stream>>>(Xc, lin1_w, lin1_b, Yb);
    predict_k<<<(MM * OO) / 256, 256, 0, stream>>>(Yb, nodes, lin2_w, lin2_b,
                                                   (float*)d_out);
}